// Col2Octree_12824772345911
// MI455X (gfx1250) — compile-verified
//
#include <hip/hip_runtime.h>

// col2octree scatter-add for MI455X (gfx1250).
//   data_in : [C=64, K=27, H=150000] f32
//   neigh   : [H=150000, K=27] int (index into N nodes, -1 = drop)
//   out     : [C=64, N=150000] f32,  out[c, neigh[h,k]] += in[c,k,h]
//
// Roofline: input stream 1.04 GB read once (~45 us @ 23.3 TB/s); output is
// 38.4 MB -> resident in the 192 MB L2, so all 259M fp32 atomics resolve at
// L2. NT loads keep the streamed input from evicting the output working set.
//
// Schedule: one thread owns one (k,h) column -> 1 index load amortized over
// 64 channel atomics. Loads are staged into registers first (one 64-wide
// coalesced NT load burst), then 64 no-return agent-scope atomics drain as
// loads complete in order.

namespace {
constexpr int C = 64;
constexpr int K = 27;
constexpr int H = 150000;
constexpr int N = 150000;
constexpr long long KH = (long long)K * H;   // channel plane stride: 4,050,000
}

__global__ __launch_bounds__(256) void Zero_f4(float4* __restrict__ out, int n4) {
  int i = blockIdx.x * blockDim.x + threadIdx.x;
  if (i < n4) out[i] = float4{0.f, 0.f, 0.f, 0.f};
}

__global__ __launch_bounds__(256) void Col2Octree_scatter(
    const float* __restrict__ data_in,
    const int*   __restrict__ neigh,
    float*       __restrict__ out) {
  const int k = blockIdx.y;                             // 0..K-1
  const int h = blockIdx.x * blockDim.x + threadIdx.x;  // lanes consecutive in h
  if (h >= H) return;

  // Index for this column; -1 marks an empty neighbor slot.
  const int idx = neigh[(long long)h * K + k];

  const float* __restrict__ src = data_in + (long long)k * H + h; // (c=0,k,h)

  // Keep the streaming input ahead in L2 (emits global_prefetch_b8).
  __builtin_prefetch(src + 1024, 0, 0);

  if (idx < 0) return;
  float* __restrict__ dst = out + idx;                            // (c=0,idx)

  // Phase 1: fire all 64 coalesced NT loads back-to-back (fills the memory
  // pipe; loads return in order).
  float v[C];
#pragma unroll
  for (int c = 0; c < C; ++c) {
    v[c] = __builtin_nontemporal_load(src + (long long)c * KH);
  }

  // Phase 2: drain with no-return agent-scope global_atomic_add_f32 into the
  // L2-resident output; each atomic only needs loadcnt <= (63-c).
#pragma unroll
  for (int c = 0; c < C; ++c) {
    __hip_atomic_fetch_add(dst + (long long)c * N, v[c],
                           __ATOMIC_RELAXED, __HIP_MEMORY_SCOPE_AGENT);
  }
}

extern "C" void kernel_launch(void* const* d_in, const int* in_sizes, int n_in,
                              void* d_out, int out_size, void* d_ws, size_t ws_size,
                              hipStream_t stream) {
  const float* data_in = (const float*)d_in[0];
  const int*   neigh   = (const int*)d_in[1];
  float*       out     = (float*)d_out;

  // Harness poisons d_out; zero it every call (deterministic, capture-safe).
  const int n4 = out_size / 4;  // C*N = 9,600,000 is divisible by 4
  Zero_f4<<<(n4 + 255) / 256, 256, 0, stream>>>((float4*)out, n4);

  dim3 grid((H + 255) / 256, K, 1);
  Col2Octree_scatter<<<grid, 256, 0, stream>>>(data_in, neigh, out);
}